// Model_90005334655100
// MI455X (gfx1250) — compile-verified
//
#include <hip/hip_runtime.h>
#include <math.h>

typedef float v2f __attribute__((ext_vector_type(2)));
typedef float v8f __attribute__((ext_vector_type(8)));

#define BT_M 128
#define BT_N 64
#define BT_K 32

// CDNA5 async global->LDS copy (ASYNCcnt-tracked), 16B per lane.
__device__ __forceinline__ void async_b128(unsigned lds_off, const float* g)
{
    asm volatile("global_load_async_to_lds_b128 %0, %1, off"
                 :: "v"(lds_off), "v"(g) : "memory");
}

// ---------------------------------------------------------------------------
// fp32 WMMA GEMM:  C[M,N] = A[M,K] * op(B) (+bias)(+relu)
//   transB==0 -> B is [K,N] row-major ; transB==1 -> B is [N,K] row-major.
//   Batched via blockIdx.z: z1=z/n2, z2=z%n2, offsets z1*s?1+z2*s?2.
//   Requires M%128==0, N%64==0, K%32==0 (true for all calls in this model).
//   Double-buffered LDS filled by global_load_async_to_lds_b128; each thread
//   issues exactly 6 asyncs per K-panel, so s_wait_asynccnt 6 == panel ready.
// ---------------------------------------------------------------------------
__global__ __launch_bounds__(256) void gemm_wmma_f32(
    const float* __restrict__ A, const float* __restrict__ B,
    float* __restrict__ C,
    int M, int N, int K, int lda, int ldb, int ldc, int transB,
    const float* __restrict__ bias, int relu,
    long sA1, long sA2, long sB1, long sB2, long sC1, long sC2, int n2)
{
    const int z  = blockIdx.z;
    const int z1 = z / n2;
    const int z2 = z - z1 * n2;
    A += (long)z1 * sA1 + (long)z2 * sA2;
    B += (long)z1 * sB1 + (long)z2 * sB2;
    C += (long)z1 * sC1 + (long)z2 * sC2;

    // A panel: 128 x 32, row stride 36 floats (144B: 16B-aligned, bank-safe)
    // B panel: notrans [32][68] (2176) or trans-staged [64][36] (2304)
    __shared__ __align__(16) float As[2][128 * 36];
    __shared__ __align__(16) float Bs[2][2304];

    const int tid  = threadIdx.x;
    const int lane = tid & 31;
    const int wave = tid >> 5;                 // 0..7 -> 16 M-rows each

    const int tileM = blockIdx.y * BT_M;
    const int tileN = blockIdx.x * BT_N;

    auto stage = [&](int buf, int k0) {
        {   // A: 128x32, 4 x b128 per thread
            const int r0 = tid >> 3;           // 0..31
            const int c  = (tid & 7) * 4;      // 0..28
#pragma unroll
            for (int u = 0; u < 4; ++u) {
                const int r = r0 + 32 * u;
                async_b128((unsigned)(size_t)&As[buf][r * 36 + c],
                           A + (long)(tileM + r) * lda + (k0 + c));
            }
        }
        if (!transB) {                          // B: 32x64, 2 x b128 per thread
            const int r0 = tid >> 4;           // 0..15
            const int c  = (tid & 15) * 4;     // 0..60
#pragma unroll
            for (int u = 0; u < 2; ++u) {
                const int r = r0 + 16 * u;
                async_b128((unsigned)(size_t)&Bs[buf][r * 68 + c],
                           B + (long)(k0 + r) * ldb + (tileN + c));
            }
        } else {                                // Bt rows staged as-is: 64x32
            const int n0 = tid >> 3;           // 0..31
            const int c  = (tid & 7) * 4;      // 0..28
#pragma unroll
            for (int u = 0; u < 2; ++u) {
                const int n = n0 + 32 * u;
                async_b128((unsigned)(size_t)&Bs[buf][n * 36 + c],
                           B + (long)(tileN + n) * ldb + (k0 + c));
            }
        }
    };

    const int m    = lane & 15;                // fragment M (A) / N (B) index
    const int kh   = lane >> 4;                // K-pair selector
    const int arow = (wave * 16 + m) * 36;

    v8f acc[4] = {};

    stage(0, 0);
    int buf = 0;
    for (int k0 = 0; k0 < K; k0 += BT_K) {
        const bool hasNext = (k0 + BT_K) < K;
        if (hasNext) {
            stage(buf ^ 1, k0 + BT_K);
            asm volatile("s_wait_asynccnt 0x6" ::: "memory");
        } else {
            asm volatile("s_wait_asynccnt 0x0" ::: "memory");
        }
        __syncthreads();

        const float* Ab = &As[buf][0];
        const float* Bb = &Bs[buf][0];
        if (!transB) {
#pragma unroll
            for (int kk = 0; kk < BT_K; kk += 4) {
                const int kq = kk + 2 * kh;
                v2f a; a.x = Ab[arow + kq]; a.y = Ab[arow + kq + 1];
#pragma unroll
                for (int cg = 0; cg < 4; ++cg) {
                    const int col = cg * 16 + m;
                    v2f b; b.x = Bb[kq * 68 + col]; b.y = Bb[(kq + 1) * 68 + col];
                    acc[cg] = __builtin_amdgcn_wmma_f32_16x16x4_f32(
                        false, a, false, b, (short)0, acc[cg], false, false);
                }
            }
        } else {
#pragma unroll
            for (int kk = 0; kk < BT_K; kk += 4) {
                const int kq = kk + 2 * kh;
                v2f a; a.x = Ab[arow + kq]; a.y = Ab[arow + kq + 1];
#pragma unroll
                for (int cg = 0; cg < 4; ++cg) {
                    const int col = cg * 16 + m;
                    v2f b; b.x = Bb[col * 36 + kq]; b.y = Bb[col * 36 + kq + 1];
                    acc[cg] = __builtin_amdgcn_wmma_f32_16x16x4_f32(
                        false, a, false, b, (short)0, acc[cg], false, false);
                }
            }
        }
        __syncthreads();
        buf ^= 1;
    }

    // writeback: VGPR vv -> row = (lane>>4)*8 + vv, col = lane&15 (per tile)
    const int rowBase = tileM + wave * 16 + kh * 8;
    const int colB0   = tileN + m;
#pragma unroll
    for (int cg = 0; cg < 4; ++cg) {
        const int col = colB0 + cg * 16;
        const float bv = bias ? bias[col] : 0.f;
#pragma unroll
        for (int vv = 0; vv < 8; ++vv) {
            float x = acc[cg][vv] + bv;
            if (relu) x = fmaxf(x, 0.f);
            C[(long)(rowBase + vv) * ldc + col] = x;
        }
    }
}

// ---------------------------------------------------------------------------
// Helpers / elementwise kernels
// ---------------------------------------------------------------------------
__device__ inline float wave_sum(float x) {
#pragma unroll
    for (int m = 16; m >= 1; m >>= 1) x += __shfl_xor(x, m, 32);
    return x;
}
__device__ inline float wave_max(float x) {
#pragma unroll
    for (int m = 16; m >= 1; m >>= 1) x = fmaxf(x, __shfl_xor(x, m, 32));
    return x;
}

// h[t, :] = embedding[inp_k[t], :]      (t = s*B + b, 2048*768 elems)
__global__ __launch_bounds__(256) void gather_embed(
    const int* __restrict__ inp_k, const float* __restrict__ emb,
    float* __restrict__ h)
{
    const long idx = (long)blockIdx.x * 256 + threadIdx.x;
    const int t = (int)(idx / 768);
    const int d = (int)(idx % 768);
    h[idx] = emb[(long)inp_k[t] * 768 + d];
}

// r2d[p, d]: p in [0,1024), pos = 512 - p ; pe = [sin | cos]
__global__ __launch_bounds__(256) void pos_emb_kernel(float* __restrict__ r2d)
{
    const long idx = (long)blockIdx.x * 256 + threadIdx.x;  // 1024*768
    const int p = (int)(idx / 768);
    const int d = (int)(idx % 768);
    const float pos = (float)(512 - p);
    const int dd = (d < 384) ? d : d - 384;
    const float inv_freq = powf(10000.0f, -(float)(2 * dd) / 768.0f);
    const float ang = pos * inv_freq;
    r2d[idx] = (d < 384) ? sinf(ang) : cosf(ang);
}

// qw = q + r_w_bias[l], qr = q + r_r_bias[l]   (broadcast over t)
__global__ __launch_bounds__(256) void add_bias2(
    const float* __restrict__ q, const float* __restrict__ rwb,
    const float* __restrict__ rrb,
    float* __restrict__ qw, float* __restrict__ qr)
{
    const long idx = (long)blockIdx.x * 256 + threadIdx.x;  // 2048*768
    const int col = (int)(idx % 768);
    const float qv = q[idx];
    qw[idx] = qv + rwb[col];
    qr[idx] = qv + rrb[col];
}

// ef[t, n, s] = dot(q[t, n*64:] + r_s_bias[l,n,:], seg_embed[l, s, n, :])
__global__ __launch_bounds__(256) void compute_ef(
    const float* __restrict__ q, const float* __restrict__ rsb,
    const float* __restrict__ seg_l, float* __restrict__ ef)
{
    const int gw   = blockIdx.x * 8 + (threadIdx.x >> 5);   // 24576 waves
    const int lane = threadIdx.x & 31;
    const int t = gw / 12;
    const int n = gw - t * 12;
    const float* qrow = q + (long)t * 768 + n * 64;
    const float* brow = rsb + n * 64;
    float p0 = 0.f, p1 = 0.f;
#pragma unroll
    for (int u = 0; u < 2; ++u) {
        const int d = lane + u * 32;
        const float qv = qrow[d] + brow[d];
        p0 += qv * seg_l[n * 64 + d];        // s = 0
        p1 += qv * seg_l[768 + n * 64 + d];  // s = 1
    }
    p0 = wave_sum(p0);
    p1 = wave_sum(p1);
    if (lane == 0) {
        ef[((long)t * 12 + n) * 2 + 0] = p0;
        ef[((long)t * 12 + n) * 2 + 1] = p1;
    }
}

// fused: score = (ac + rel_shift(bd) + ef[diff]) * scale, causal mask,
// softmax over j. prob written in-place over ac. one wave per (z, i) row.
__global__ __launch_bounds__(256) void attn_softmax(
    const float* __restrict__ ac, const float* __restrict__ bd,
    const float* __restrict__ ef, const int* __restrict__ seg_id,
    float* __restrict__ prob, float scale)
{
    const int gw   = blockIdx.x * 8 + (threadIdx.x >> 5);  // 48*512 waves
    const int lane = threadIdx.x & 31;
    const int z = gw >> 9;       // (b,n) batch index
    const int i = gw & 511;
    const int b = z / 12;
    const int n = z - b * 12;

    const float* acr = ac + ((long)z * 512 + i) * 512;
    const float* bdr = bd + ((long)z * 512 + i) * 1024;
    float*       prr = prob + ((long)z * 512 + i) * 512;

    const long efo = ((long)(i * 4 + b) * 12 + n) * 2;
    const float e0 = ef[efo + 0];
    const float e1 = ef[efo + 1];
    const int   si = seg_id[i * 4 + b];

    float sc[16];
    float mx = -3.0e38f;
#pragma unroll
    for (int t = 0; t < 16; ++t) {
        const int j = t * 32 + lane;
        float s = acr[j] + bdr[j - i + 512];           // rel_shift gather
        s += (seg_id[j * 4 + b] != si) ? e1 : e0;      // segment one-hot
        s *= scale;
        if (j > i) s = -1.0e30f;                       // causal mask
        sc[t] = s;
        mx = fmaxf(mx, s);
    }
    mx = wave_max(mx);
    float sum = 0.f;
#pragma unroll
    for (int t = 0; t < 16; ++t) {
        const float e = expf(sc[t] - mx);
        sc[t] = e;
        sum += e;
    }
    sum = wave_sum(sum);
    const float inv = 1.0f / sum;
#pragma unroll
    for (int t = 0; t < 16; ++t) prr[t * 32 + lane] = sc[t] * inv;
}

// h = LayerNorm(x + h) * g + b    (one wave per 768-wide row; 2048 rows)
__global__ __launch_bounds__(256) void add_layernorm(
    const float* __restrict__ x, float* __restrict__ h,
    const float* __restrict__ g, const float* __restrict__ bta)
{
    const int row  = blockIdx.x * 8 + (threadIdx.x >> 5);
    const int lane = threadIdx.x & 31;
    const float* xr = x + (long)row * 768;
    float*       hr = h + (long)row * 768;

    float v[24];
    float s = 0.f;
#pragma unroll
    for (int t = 0; t < 24; ++t) {
        const int c = t * 32 + lane;
        v[t] = xr[c] + hr[c];
        s += v[t];
    }
    const float mean = wave_sum(s) * (1.0f / 768.0f);
    float var = 0.f;
#pragma unroll
    for (int t = 0; t < 24; ++t) {
        const float d = v[t] - mean;
        var += d * d;
    }
    var = wave_sum(var) * (1.0f / 768.0f);
    const float inv = rsqrtf(var + 1e-5f);
#pragma unroll
    for (int t = 0; t < 24; ++t) {
        const int c = t * 32 + lane;
        hr[c] = (v[t] - mean) * inv * g[c] + bta[c];
    }
}

// ---------------------------------------------------------------------------
static inline void launch_gemm(hipStream_t s, const float* A, const float* B,
                               float* C, int M, int N, int K,
                               int lda, int ldb, int ldc, int transB,
                               const float* bias, int relu,
                               long sA1, long sA2, long sB1, long sB2,
                               long sC1, long sC2, int n2, int nb)
{
    dim3 grid(N / BT_N, M / BT_M, nb);
    gemm_wmma_f32<<<grid, dim3(256), 0, s>>>(A, B, C, M, N, K, lda, ldb, ldc,
                                             transB, bias, relu,
                                             sA1, sA2, sB1, sB2, sC1, sC2, n2);
}

extern "C" void kernel_launch(void* const* d_in, const int* in_sizes, int n_in,
                              void* d_out, int out_size, void* d_ws, size_t ws_size,
                              hipStream_t stream)
{
    (void)in_sizes; (void)n_in; (void)out_size; (void)ws_size;

    const int*   inp_k     = (const int*)  d_in[0];
    const int*   seg_id    = (const int*)  d_in[1];
    const float* emb       = (const float*)d_in[2];
    const float* Wq        = (const float*)d_in[3];
    const float* Wk        = (const float*)d_in[4];
    const float* Wv        = (const float*)d_in[5];
    const float* Wr        = (const float*)d_in[6];
    const float* proj_o    = (const float*)d_in[7];
    const float* r_w_bias  = (const float*)d_in[8];
    const float* r_r_bias  = (const float*)d_in[9];
    const float* r_s_bias  = (const float*)d_in[10];
    const float* seg_embed = (const float*)d_in[11];
    const float* ln_g      = (const float*)d_in[12];
    const float* ln_b      = (const float*)d_in[13];
    const float* W1        = (const float*)d_in[14];
    const float* b1        = (const float*)d_in[15];
    const float* W2        = (const float*)d_in[16];
    const float* b2        = (const float*)d_in[17];
    const float* softmax_b = (const float*)d_in[18];
    float* out = (float*)d_out;

    const long TD = 2048l * 768;
    float* ws   = (float*)d_ws;
    float* h    = ws; ws += TD;
    float* r2d  = ws; ws += 1024l * 768;
    float* kr2d = ws; ws += 1024l * 768;
    float* q    = ws; ws += TD;
    float* k    = ws; ws += TD;
    float* v    = ws; ws += TD;
    float* qw   = ws; ws += TD;           // reused as attn_out
    float* qr   = ws; ws += TD;           // reused as ffn output f2
    float* avec = ws; ws += TD;
    float* ef   = ws; ws += 2048l * 12 * 2;
    float* ac   = ws; ws += 48l * 512 * 512;   // reused as prob (in-place)
    float* bd   = ws; ws += 48l * 512 * 1024;  // reused as ffn hidden a1
    float* attn_out = qw;
    float* a1 = bd;
    float* f2 = qr;

    // h = embedding[inp_k] ; r = pos_emb (batch-independent)
    gather_embed<<<6144, 256, 0, stream>>>(inp_k, emb, h);
    pos_emb_kernel<<<3072, 256, 0, stream>>>(r2d);

    // kr = r @ Wr  (layer- and batch-independent: compute once)
    launch_gemm(stream, r2d, Wr, kr2d, 1024, 768, 768, 768, 768, 768, 0,
                nullptr, 0, 0, 0, 0, 0, 0, 0, 1, 1);

    const float scale = 0.125f;  // 1/sqrt(64)

    for (int l = 0; l < 4; ++l) {
        // q/k/v projections: [2048,768] @ [768,768]
        launch_gemm(stream, h, Wq, q, 2048, 768, 768, 768, 768, 768, 0,
                    nullptr, 0, 0, 0, 0, 0, 0, 0, 1, 1);
        launch_gemm(stream, h, Wk, k, 2048, 768, 768, 768, 768, 768, 0,
                    nullptr, 0, 0, 0, 0, 0, 0, 0, 1, 1);
        launch_gemm(stream, h, Wv, v, 2048, 768, 768, 768, 768, 768, 0,
                    nullptr, 0, 0, 0, 0, 0, 0, 0, 1, 1);

        add_bias2<<<6144, 256, 0, stream>>>(q, r_w_bias + l * 768,
                                            r_r_bias + l * 768, qw, qr);
        compute_ef<<<3072, 256, 0, stream>>>(q, r_s_bias + l * 768,
                                             seg_embed + (long)l * 1536, ef);

        // ac[z=b*12+n] = (q+rwb)(512x64) @ k^T(64x512)
        launch_gemm(stream, qw, k, ac, 512, 512, 64, 3072, 3072, 512, 1,
                    nullptr, 0,
                    768, 64, 768, 64, 12l * 512 * 512, 512l * 512, 12, 48);
        // bd_full[z] = (q+rrb)(512x64) @ kr^T(64x1024)
        launch_gemm(stream, qr, kr2d, bd, 512, 1024, 64, 3072, 768, 1024, 1,
                    nullptr, 0,
                    768, 64, 0, 64, 12l * 512 * 1024, 512l * 1024, 12, 48);

        // fused rel-shift + segment + mask + softmax (prob in-place on ac)
        attn_softmax<<<3072, 256, 0, stream>>>(ac, bd, ef, seg_id, ac, scale);

        // attn_vec[z] = prob(512x512) @ v(512x64)
        launch_gemm(stream, ac, v, avec, 512, 64, 512, 512, 3072, 3072, 0,
                    nullptr, 0,
                    12l * 512 * 512, 512l * 512, 768, 64, 768, 64, 12, 48);

        // attn_out = attn_vec(2048x768) @ proj_o^T
        launch_gemm(stream, avec, proj_o, attn_out, 2048, 768, 768,
                    768, 768, 768, 1, nullptr, 0, 0, 0, 0, 0, 0, 0, 1, 1);
        add_layernorm<<<256, 256, 0, stream>>>(attn_out, h, ln_g, ln_b);

        // FFN: relu(h @ W1 + b1) @ W2 + b2
        launch_gemm(stream, h, W1, a1, 2048, 3072, 768, 768, 3072, 3072, 0,
                    b1, 1, 0, 0, 0, 0, 0, 0, 1, 1);
        launch_gemm(stream, a1, W2, f2, 2048, 768, 3072, 3072, 768, 768, 0,
                    b2, 0, 0, 0, 0, 0, 0, 0, 1, 1);
        add_layernorm<<<256, 256, 0, stream>>>(f2, h, ln_g, ln_b);
    }

    // logits = h @ embedding^T + softmax_b   -> [2048, 32000] = [S,B,V]
    launch_gemm(stream, h, emb, out, 2048, 32000, 768, 768, 768, 32000, 1,
                softmax_b, 0, 0, 0, 0, 0, 0, 0, 1, 1);
}